// LatentChargeModel_52158082842747
// MI455X (gfx1250) — compile-verified
//
#include <hip/hip_runtime.h>

typedef float v2f __attribute__((ext_vector_type(2)));
typedef float v8f __attribute__((ext_vector_type(8)));

__device__ __forceinline__ float lane_bcast(float v, int srcLane) {
    // dst[lane] = v[srcLane]; byte index for ds_bpermute
    return __int_as_float(__builtin_amdgcn_ds_bpermute(srcLane << 2, __float_as_int(v)));
}

// One BLOCK (8 waves) handles one (batch, 16-atom j-tile); each wave covers
// NT/8 i-tiles, partials combined deterministically through LDS.
// Lane owns column atom j = jt*16 + (lane&15); accumulates forces[j] in-lane.
// WMMA #1: C[m][n] = -2*p_i.p_j + |p_i|^2 via the K=4 slot:
//   A = [x_i, y_i, z_i, ri2], B = [-2x_j, -2y_j, -2z_j, 1]  =>  d2 = rj2 + C
// WMMA #2: charge outer product qq[m][n] = q_i * q_j (rank-1).
__global__ __launch_bounds__(256) void coulomb_wmma_kernel(
    const float* __restrict__ pos,   // [B,N,3]
    const float* __restrict__ q,     // [N]
    float* __restrict__ forces,      // [B,N,3] (d_out + B)
    float* __restrict__ e_ws,        // [B*NT] per-block energy partials
    int N, int NT)
{
    const int lane = threadIdx.x & 31;
    const int w    = threadIdx.x >> 5;          // wave index in block: 0..7
    const int blk  = blockIdx.x;                // one block per (b, jt)
    const int b    = blk / NT;
    const int jt   = blk - b * NT;
    const int half = lane >> 4;                 // 0: rows 0..7 / K 0..1, 1: rows 8..15 / K 2..3
    const int n    = lane & 15;                 // column within tile

    const float* posb = pos + (size_t)b * N * 3;

    // ---- column atom (fixed for the whole block) ----
    const int   j   = jt * 16 + n;
    const float xb  = posb[j * 3 + 0];
    const float yb  = posb[j * 3 + 1];
    const float zb  = posb[j * 3 + 2];
    const float qb  = q[j];
    const float rj2 = xb * xb + yb * yb + zb * zb;

    // B matrix 4x16 (KxN), per-lane: vgpr0 = K(2*half), vgpr1 = K(2*half+1)
    v2f bm;                                     // [-2x_j, -2y_j, -2z_j, 1]
    bm.x = half ? (-2.0f * zb) : (-2.0f * xb);
    bm.y = half ? 1.0f         : (-2.0f * yb);
    v2f bq;                                     // [q_j, 0, 0, 0]
    bq.x = half ? 0.0f : qb;
    bq.y = 0.0f;

    float e = 0.0f, fx = 0.0f, fy = 0.0f, fz = 0.0f;

    const int tpw = (NT + 7) >> 3;              // i-tiles per wave
    const int it0 = w * tpw;
    const int it1 = (it0 + tpw < NT) ? (it0 + tpw) : NT;

    for (int it = it0; it < it1; ++it) {
        // stage row-tile atom (lane&15) for broadcasts
        const int   ia  = it * 16 + n;
        const float xa  = posb[ia * 3 + 0];
        const float ya  = posb[ia * 3 + 1];
        const float za  = posb[ia * 3 + 2];
        const float qa  = q[ia];
        const float ra2 = xa * xa + ya * ya + za * za;

        // A matrix 16x4 (MxK): [x_i, y_i, z_i, ri2]
        v2f am;
        am.x = half ? za  : xa;
        am.y = half ? ra2 : ya;
        v2f aq;                                 // [q_i, 0, 0, 0]
        aq.x = half ? 0.0f : qa;
        aq.y = 0.0f;

        v8f c  = {};
        v8f c2 = {};
        c  = __builtin_amdgcn_wmma_f32_16x16x4_f32(
                 false, am, false, bm, (short)0, c,  false, false);
        c2 = __builtin_amdgcn_wmma_f32_16x16x4_f32(
                 false, aq, false, bq, (short)0, c2, false, false);

        const bool tile_diag = (it == jt);
        #pragma unroll
        for (int r = 0; r < 8; ++r) {
            const int m = r + 8 * half;           // row of this C vgpr for this lane half

            float d2 = fmaxf(rj2 + c[r], 1e-12f); // |pi-pj|^2 (guard cancellation)
            float inv_r = __frsqrt_rn(d2);
            if (tile_diag && (m == n)) inv_r = 0.0f;   // i == j masked out

            const float qq = c2[r];               // q_i * q_j from WMMA
            const float ee = qq * inv_r;
            e += ee;
            const float s = ee * inv_r * inv_r;   // qq / r^3 (symmetric in i,j)

            const float xi = lane_bcast(xa, m);
            const float yi = lane_bcast(ya, m);
            const float zi = lane_bcast(za, m);
            fx = fmaf(s, xi - xb, fx);            // contribution to forces[j]
            fy = fmaf(s, yi - yb, fy);
            fz = fmaf(s, zi - zb, fz);
        }
    }

    // combine the two row-halves (lanes l and l^16 hold the same column j)
    fx += lane_bcast(fx, lane ^ 16);
    fy += lane_bcast(fy, lane ^ 16);
    fz += lane_bcast(fz, lane ^ 16);

    // wave-wide energy reduction (fixed order -> deterministic)
    #pragma unroll
    for (int off = 1; off < 32; off <<= 1)
        e += lane_bcast(e, lane ^ off);

    // ---- deterministic cross-wave combine through LDS ----
    __shared__ float smx[8][16];
    __shared__ float smy[8][16];
    __shared__ float smz[8][16];
    __shared__ float sme[8];
    if (lane < 16) {
        smx[w][lane] = fx;
        smy[w][lane] = fy;
        smz[w][lane] = fz;
    }
    if (lane == 0) sme[w] = e;
    __syncthreads();

    const int t = threadIdx.x;
    if (t < 16) {
        float sfx = 0.0f, sfy = 0.0f, sfz = 0.0f;
        #pragma unroll
        for (int ww = 0; ww < 8; ++ww) {
            sfx += smx[ww][t];
            sfy += smy[ww][t];
            sfz += smz[ww][t];
        }
        float* fp = forces + ((size_t)b * N + jt * 16 + t) * 3;
        fp[0] = sfx; fp[1] = sfy; fp[2] = sfz;
    } else if (t == 16) {
        float se = 0.0f;
        #pragma unroll
        for (int ww = 0; ww < 8; ++ww) se += sme[ww];
        e_ws[b * NT + jt] = se;
    }
}

// energy[b] = 0.5 * sum_jt e_ws[b*NT + jt], fixed-order tree reduction
__global__ void coulomb_energy_reduce(const float* __restrict__ e_ws,
                                      float* __restrict__ energy, int NT)
{
    __shared__ float sm[128];
    const int b = blockIdx.x;
    const int t = threadIdx.x;
    sm[t] = (t < NT) ? e_ws[b * NT + t] : 0.0f;
    __syncthreads();
    for (int off = 64; off > 0; off >>= 1) {
        if (t < off) sm[t] += sm[t + off];
        __syncthreads();
    }
    if (t == 0) energy[b] = 0.5f * sm[0];
}

extern "C" void kernel_launch(void* const* d_in, const int* in_sizes, int n_in,
                              void* d_out, int out_size, void* d_ws, size_t ws_size,
                              hipStream_t stream) {
    const float* pos = (const float*)d_in[0];   // [B,N,3]
    const float* q   = (const float*)d_in[1];   // [N]
    const int N  = in_sizes[1];                 // 2048
    const int B  = in_sizes[0] / (3 * N);       // 8
    const int NT = N / 16;                      // 128 tiles per dim

    float* energy = (float*)d_out;              // [B]
    float* forces = (float*)d_out + B;          // [B,N,3]
    float* e_ws   = (float*)d_ws;               // [B*NT] floats

    const int blocks = B * NT;                  // 1024 blocks, 8 waves each
    coulomb_wmma_kernel<<<blocks, 256, 0, stream>>>(pos, q, forces, e_ws, N, NT);
    coulomb_energy_reduce<<<B, 128, 0, stream>>>(e_ws, energy, NT);
}